// KernelRegression_40243843564035
// MI455X (gfx1250) — compile-verified
//
#include <hip/hip_runtime.h>

typedef __attribute__((ext_vector_type(2))) float v2f;
typedef __attribute__((ext_vector_type(8))) float v8f;

#define MM   64   // inducing points
#define OUTC 2    // output columns

// ---------------------------------------------------------------------------
// Setup kernel (1 block, 64 threads): Kzz -> Cholesky L -> solve L^T W = U
// (since inv(Kzz)@L == L^{-T}), write W padded to [64][16] (cols>=2 zero).
// ---------------------------------------------------------------------------
__global__ __launch_bounds__(64) void gp_setup(const float* __restrict__ Z,
                                               const float* __restrict__ U,
                                               const float* __restrict__ sf_p,
                                               const float* __restrict__ ell_p,
                                               float* __restrict__ Wpad) {
  __shared__ float A[MM * MM];   // Kzz, then Cholesky factor (lower) in place
  __shared__ float Zs[MM * 3];
  __shared__ float Wsh[MM * OUTC];

  const int tid = threadIdx.x;            // 0..63
  const float sf  = sf_p[0];
  const float iel = 1.0f / ell_p[0];
  const float sf2 = sf * sf;

  // scaled inducing points
  const float zx = Z[tid * 3 + 0] * iel;
  const float zy = Z[tid * 3 + 1] * iel;
  const float zz = Z[tid * 3 + 2] * iel;
  Zs[tid * 3 + 0] = zx; Zs[tid * 3 + 1] = zy; Zs[tid * 3 + 2] = zz;
  __syncthreads();

  // Kzz row `tid`
  for (int j = 0; j < MM; ++j) {
    const float dx = zx - Zs[j * 3 + 0];
    const float dy = zy - Zs[j * 3 + 1];
    const float dz = zz - Zs[j * 3 + 2];
    const float sq = dx * dx + dy * dy + dz * dz;
    A[tid * MM + j] = sf2 * __expf(-0.5f * sq);
  }
  __syncthreads();

  // Cooperative Cholesky (lower triangle), thread tid owns row tid
  for (int k = 0; k < MM; ++k) {
    if (tid == k) A[k * MM + k] = sqrtf(A[k * MM + k]);
    __syncthreads();
    if (tid > k) A[tid * MM + k] /= A[k * MM + k];
    __syncthreads();
    if (tid > k) {
      const float lik = A[tid * MM + k];
      for (int j = k + 1; j <= tid; ++j)
        A[tid * MM + j] -= lik * A[j * MM + k];
    }
    __syncthreads();
  }

  // Back-substitution: L^T W = U  (threads 0,1 each own one RHS column)
  if (tid < OUTC) {
    for (int i = MM - 1; i >= 0; --i) {
      float s = U[i * OUTC + tid];
      for (int j = i + 1; j < MM; ++j)
        s -= A[j * MM + i] * Wsh[j * OUTC + tid];
      Wsh[i * OUTC + tid] = s / A[i * MM + i];
    }
  }
  __syncthreads();

  // Write W padded to [64][16] (WMMA B operand; cols 2..15 are zero)
  for (int idx = tid; idx < MM * 16; idx += 64) {
    const int r = idx >> 4, c = idx & 15;
    Wpad[idx] = (c < OUTC) ? Wsh[r * OUTC + c] : 0.0f;
  }
}

// ---------------------------------------------------------------------------
// Main kernel: each wave computes a 16-row tile of Kxz directly in the
// V_WMMA_F32_16X16X4_F32 A-fragment layout and accumulates out = Kxz @ Wpad.
// ---------------------------------------------------------------------------
__global__ __launch_bounds__(256) void gp_forward(const float* __restrict__ X,
                                                  const float* __restrict__ Z,
                                                  const float* __restrict__ sf_p,
                                                  const float* __restrict__ ell_p,
                                                  const float* __restrict__ Wpad,
                                                  float* __restrict__ out,
                                                  int n, int numTiles) {
  __shared__ float zs[MM * 4];   // z'/ell (x,y,z) and 0.5*|z'|^2 per column

  const int tid  = threadIdx.x;
  const int lane = tid & 31;
  const int wIdx = tid >> 5;           // wave in block (blockDim=256 -> 8)
  const int hlf  = lane >> 4;          // 0: lanes 0-15, 1: lanes 16-31
  const int nIdx = lane & 15;          // A: row-in-tile; B/D: column
  const int koff = hlf * 2;            // K-local offset within each K=4 step

  const float iel = 1.0f / ell_p[0];
  const float sf2 = sf_p[0] * sf_p[0];

  if (tid < MM) {
    const float a = Z[tid * 3 + 0] * iel;
    const float b = Z[tid * 3 + 1] * iel;
    const float c = Z[tid * 3 + 2] * iel;
    zs[tid * 4 + 0] = a; zs[tid * 4 + 1] = b; zs[tid * 4 + 2] = c;
    zs[tid * 4 + 3] = 0.5f * (a * a + b * b + c * c);
  }

  // B fragments (4x16 f32): VGPR0 = K-local {0|2}, VGPR1 = K-local {1|3}
  v2f bfrag[16];
#pragma unroll
  for (int kk = 0; kk < 16; ++kk) {
    bfrag[kk].x = Wpad[(4 * kk + koff)     * 16 + nIdx];
    bfrag[kk].y = Wpad[(4 * kk + koff + 1) * 16 + nIdx];
  }
  __syncthreads();

  const int wavesTotal = gridDim.x * 8;
  const int wave = blockIdx.x * 8 + wIdx;

  for (int tile = wave; tile < numTiles; tile += wavesTotal) {
    const int row = tile * 16 + nIdx;
    const int rc  = row < n ? row : n - 1;       // clamp (N%16==0 anyway)
    const float x0 = X[rc * 3 + 0] * iel;
    const float x1 = X[rc * 3 + 1] * iel;
    const float x2 = X[rc * 3 + 2] * iel;
    const float xn = 0.5f * (x0 * x0 + x1 * x1 + x2 * x2);

    v8f acc = {};
#pragma unroll
    for (int kk = 0; kk < 16; ++kk) {
      const int c0 = 4 * kk + koff;              // this lane's two K columns
      const float e0 = xn + zs[c0 * 4 + 3]
                     - (x0 * zs[c0 * 4 + 0] + x1 * zs[c0 * 4 + 1] + x2 * zs[c0 * 4 + 2]);
      const float e1 = xn + zs[(c0 + 1) * 4 + 3]
                     - (x0 * zs[(c0 + 1) * 4 + 0] + x1 * zs[(c0 + 1) * 4 + 1] + x2 * zs[(c0 + 1) * 4 + 2]);
      v2f afrag;
      afrag.x = sf2 * __expf(-e0);
      afrag.y = sf2 * __expf(-e1);
      // D = A(16x4 f32) x B(4x16 f32) + C   (full f32 precision)
      acc = __builtin_amdgcn_wmma_f32_16x16x4_f32(
          /*neg_a=*/false, afrag, /*neg_b=*/false, bfrag[kk],
          /*c_mod=*/(short)0, acc, /*reuse_a=*/false, /*reuse_b=*/false);
    }

    // D layout: lane n = col, VGPR v -> row v + 8*hlf. Only cols 0,1 are real.
    if (nIdx < OUTC) {
      const int base = tile * 16 + hlf * 8;
#pragma unroll
      for (int v = 0; v < 8; ++v) {
        const int r = base + v;
        if (r < n) out[r * OUTC + nIdx] = acc[v];
      }
    }
  }
}

// ---------------------------------------------------------------------------
extern "C" void kernel_launch(void* const* d_in, const int* in_sizes, int n_in,
                              void* d_out, int out_size, void* d_ws, size_t ws_size,
                              hipStream_t stream) {
  // setup_inputs order: t, X, Z, U, sf, ell
  const float* X   = (const float*)d_in[1];
  const float* Z   = (const float*)d_in[2];
  const float* U   = (const float*)d_in[3];
  const float* sf  = (const float*)d_in[4];
  const float* ell = (const float*)d_in[5];
  float* out  = (float*)d_out;
  float* Wpad = (float*)d_ws;            // 64*16 floats = 4 KB scratch

  const int n        = in_sizes[1] / 3;  // N from X
  const int numTiles = (n + 15) / 16;

  gp_setup<<<1, 64, 0, stream>>>(Z, U, sf, ell, Wpad);

  int blocks = 2048;                      // persistent waves, grid-stride tiles
  const int maxBlocks = (numTiles + 7) / 8;
  if (blocks > maxBlocks) blocks = maxBlocks;
  gp_forward<<<blocks, 256, 0, stream>>>(X, Z, sf, ell, Wpad, out, n, numTiles);
}